// SpatialTransformer_48558900248899
// MI455X (gfx1250) — compile-verified
//
#include <hip/hip_runtime.h>
#include <stdint.h>

// SpatialTransformer (flow warp, bilinear, zeros padding) for MI455X/gfx1250.
// Memory-bound gather: ~270-400 MB HBM traffic -> ~12-17 us at 23.3 TB/s.
// No matmul structure -> no WMMA; uses gfx1250 async-to-LDS + NT stores + prefetch.

#define BDIM 256

__global__ __launch_bounds__(BDIM) void warp_bilinear_gfx1250(
    const float* __restrict__ src,   // [B,C,H,W]
    const float* __restrict__ flow,  // [B,2,H,W]
    float* __restrict__ out)         // [B,C,H,W]
{
    constexpr int B = 8, C = 64, H = 256, W = 256;
    constexpr int HW = H * W;
    (void)B;

    const int x = threadIdx.x;   // one thread per pixel column
    const int y = blockIdx.x;    // one block per row
    const int b = blockIdx.y;    // batch

    // ---- Stage the flow tile into LDS via the CDNA5 async-to-LDS path ----
    __shared__ float sflow[2 * BDIM];

    const float* f0p = flow + ((size_t)(b * 2 + 0) * HW + (size_t)y * W + x);
    const float* f1p = f0p + HW;

    // LDS flat addresses: aperture lives in the upper 32 bits, low 32 bits
    // are the LDS byte offset (ISA 10.2 aperture mapping).
    unsigned lds0 = (unsigned)(uintptr_t)&sflow[x];
    unsigned lds1 = (unsigned)(uintptr_t)&sflow[BDIM + x];

    asm volatile("global_load_async_to_lds_b32 %0, %1, off"
                 :: "v"(lds0), "v"(f0p) : "memory");
    asm volatile("global_load_async_to_lds_b32 %0, %1, off"
                 :: "v"(lds1), "v"(f1p) : "memory");
    asm volatile("s_wait_asynccnt 0x0" ::: "memory");

    float f0 = sflow[x];          // flow along H (y)
    float f1 = sflow[BDIM + x];   // flow along W (x)

    // ---- Per-pixel tap setup (amortized over 64 channels) ----
    float py = (float)y + f0;
    float px = (float)x + f1;
    // Reference does a normalize->denormalize round trip (identity up to
    // rounding); replicate it for bit-level fidelity.
    constexpr float invHm1 = 1.0f / (float)(H - 1);
    constexpr float invWm1 = 1.0f / (float)(W - 1);
    py = (2.0f * (py * invHm1 - 0.5f) + 1.0f) * 0.5f * (float)(H - 1);
    px = (2.0f * (px * invWm1 - 0.5f) + 1.0f) * 0.5f * (float)(W - 1);

    float y0f = floorf(py), x0f = floorf(px);
    float wy1 = py - y0f, wy0 = 1.0f - wy1;
    float wx1 = px - x0f, wx0 = 1.0f - wx1;

    int y0 = (int)y0f, x0i = (int)x0f;
    int y1 = y0 + 1,  x1i = x0i + 1;

    float vy0 = (y0  >= 0 && y0  < H) ? 1.0f : 0.0f;
    float vy1 = (y1  >= 0 && y1  < H) ? 1.0f : 0.0f;
    float vx0 = (x0i >= 0 && x0i < W) ? 1.0f : 0.0f;
    float vx1 = (x1i >= 0 && x1i < W) ? 1.0f : 0.0f;

    int yc0 = min(max(y0, 0), H - 1), yc1 = min(max(y1, 0), H - 1);
    int xc0 = min(max(x0i, 0), W - 1), xc1 = min(max(x1i, 0), W - 1);

    // zeros padding == weight * valid
    float w00 = wy0 * wx0 * vy0 * vx0;
    float w01 = wy0 * wx1 * vy0 * vx1;
    float w10 = wy1 * wx0 * vy1 * vx0;
    float w11 = wy1 * wx1 * vy1 * vx1;

    int o00 = yc0 * W + xc0, o01 = yc0 * W + xc1;
    int o10 = yc1 * W + xc0, o11 = yc1 * W + xc1;

    const float* sb = src + (size_t)b * C * HW;
    float*       ob = out + (size_t)b * C * HW + (size_t)y * W + x;

    // ---- Channel loop: 4 gathers + 3 FMA + 1 NT store per channel ----
    #pragma unroll 4
    for (int c = 0; c < C - 8; ++c) {
        const float* sc = sb + c * HW;
        // Warm L2 for the strided channel walk (global_prefetch_b8).
        __builtin_prefetch(sc + 8 * HW + o00, 0, 1);
        float v = fmaf(w00, sc[o00],
                  fmaf(w01, sc[o01],
                  fmaf(w10, sc[o10],
                       w11 * sc[o11])));
        // NT store: keep the 192MB L2 for src (which nearly fits entirely).
        __builtin_nontemporal_store(v, ob + (size_t)c * HW);
    }
    #pragma unroll
    for (int c = C - 8; c < C; ++c) {
        const float* sc = sb + c * HW;
        float v = fmaf(w00, sc[o00],
                  fmaf(w01, sc[o01],
                  fmaf(w10, sc[o10],
                       w11 * sc[o11])));
        __builtin_nontemporal_store(v, ob + (size_t)c * HW);
    }
}

extern "C" void kernel_launch(void* const* d_in, const int* in_sizes, int n_in,
                              void* d_out, int out_size, void* d_ws, size_t ws_size,
                              hipStream_t stream) {
    const float* src  = (const float*)d_in[0];  // [8,64,256,256] f32
    const float* flow = (const float*)d_in[1];  // [8,2,256,256]  f32
    float* out = (float*)d_out;                 // [8,64,256,256] f32
    (void)in_sizes; (void)n_in; (void)out_size; (void)d_ws; (void)ws_size;

    dim3 grid(256 /*H rows*/, 8 /*B*/);
    warp_bilinear_gfx1250<<<grid, BDIM, 0, stream>>>(src, flow, out);
}